// SSDGuaranteeLoss_50912542327367
// MI455X (gfx1250) — compile-verified
//
#include <hip/hip_runtime.h>

#define NB 32
#define NA 24564
#define NC 81
#define NM 20
#define THRESH 0.5f
#define VAR0 0.1f
#define VAR1 0.2f

typedef __attribute__((ext_vector_type(2))) float v2f;
typedef __attribute__((ext_vector_type(8))) float v8f;

__device__ __forceinline__ unsigned f2k(float f) {
  unsigned b = __float_as_uint(f);
  return (b & 0x80000000u) ? ~b : (b | 0x80000000u);
}
__device__ __forceinline__ float k2f(unsigned k) {
  unsigned b = (k & 0x80000000u) ? (k & 0x7fffffffu) : ~k;
  return __uint_as_float(b);
}

// ---------------- K0: zero-init accumulators ----------------
__global__ void k_init(unsigned long long* __restrict__ packed, int* __restrict__ pos_num) {
  int tid = blockIdx.x * blockDim.x + threadIdx.x;
  if (tid < NB * NM) packed[tid] = 0ull;
  if (tid < NB) pos_num[tid] = 0;
}

// ---------------- K1: IoU matching ----------------
// per (b,a): best over M (max + first-argmax); per (b,m): packed atomicMax over anchors
__global__ void k_match(const float* __restrict__ anchors, const float* __restrict__ targets,
                        float* __restrict__ best_iou, int* __restrict__ best_idx,
                        unsigned long long* __restrict__ packed) {
  __shared__ float st[NM * 5];
  __shared__ unsigned long long sbest[NM];
  int b = blockIdx.y;
  int tid = threadIdx.x;
  if (tid < NM * 5) st[tid] = targets[b * NM * 5 + tid];
  if (tid < NM) sbest[tid] = 0ull;
  __syncthreads();
  int a = blockIdx.x * blockDim.x + tid;
  if (a < NA) {
    float acx = anchors[a * 4 + 0], acy = anchors[a * 4 + 1];
    float aw = anchors[a * 4 + 2], ah = anchors[a * 4 + 3];
    float ax0 = acx - aw * 0.5f, ay0 = acy - ah * 0.5f;
    float ax1 = acx + aw * 0.5f, ay1 = acy + ah * 0.5f;
    float area_a = (ax1 - ax0) * (ay1 - ay0);
    float bi = -1.0f;
    int bm = 0;
    for (int m = 0; m < NM; ++m) {
      float x0 = st[m * 5 + 0], y0 = st[m * 5 + 1];
      float x1 = st[m * 5 + 2], y1 = st[m * 5 + 3];
      float lx = fmaxf(x0, ax0), ly = fmaxf(y0, ay0);
      float rx = fminf(x1, ax1), ry = fminf(y1, ay1);
      float w = fmaxf(rx - lx, 0.f), h = fmaxf(ry - ly, 0.f);
      float inter = w * h;
      float area_b = (x1 - x0) * (y1 - y0);
      float iou = inter / (area_b + area_a - inter);
      if (iou > bi) { bi = iou; bm = m; }  // strict > : first max wins over M
      // iou>=0 so float bits are monotone; ~a so ties pick smallest anchor index
      unsigned long long p =
          ((unsigned long long)__float_as_uint(iou) << 32) | (unsigned)(~(unsigned)a);
      atomicMax(&sbest[m], p);
    }
    best_iou[(size_t)b * NA + a] = bi;
    best_idx[(size_t)b * NA + a] = bm;
  }
  __syncthreads();
  if (tid < NM) atomicMax(&packed[b * NM + tid], sbest[tid]);
}

// ---------------- K2: guarantee (scatter, last m wins) ----------------
__global__ void k_guarantee(const unsigned long long* __restrict__ packed,
                            float* __restrict__ best_iou, int* __restrict__ best_idx) {
  int b = blockIdx.x * blockDim.x + threadIdx.x;
  if (b < NB) {
    for (int m = 0; m < NM; ++m) {
      unsigned a = ~(unsigned)(packed[b * NM + m] & 0xffffffffull);
      best_iou[(size_t)b * NA + a] = 2.0f;
      best_idx[(size_t)b * NA + a] = m;
    }
  }
}

// ---------------- K3: encode + SmoothL1 partials + pos counts ----------------
__global__ void k_encode(const float* __restrict__ targets, const float* __restrict__ anchors,
                         const float* __restrict__ pred_loc,
                         const float* __restrict__ best_iou, const int* __restrict__ best_idx,
                         int* __restrict__ conf_t, float* __restrict__ loc_part,
                         int* __restrict__ pos_num) {
  __shared__ float st[NM * 5];
  __shared__ float fred[256];
  __shared__ int ired[256];
  int b = blockIdx.y, tid = threadIdx.x;
  if (tid < NM * 5) st[tid] = targets[b * NM * 5 + tid];
  __syncthreads();
  int a = blockIdx.x * blockDim.x + tid;
  float lsum = 0.f;
  int lpos = 0;
  if (a < NA) {
    size_t idx = (size_t)b * NA + a;
    int gt = best_idx[idx];
    float iouv = best_iou[idx];
    float lab = st[gt * 5 + 4];
    int conf = (iouv < THRESH) ? 0 : ((int)lab + 1);
    conf_t[idx] = conf;
    if (conf > 0) {
      lpos = 1;
      float x0 = st[gt * 5 + 0], y0 = st[gt * 5 + 1];
      float x1 = st[gt * 5 + 2], y1 = st[gt * 5 + 3];
      float acx = anchors[a * 4 + 0], acy = anchors[a * 4 + 1];
      float aw = anchors[a * 4 + 2], ah = anchors[a * 4 + 3];
      float g[4];
      g[0] = ((x0 + x1) * 0.5f - acx) / (VAR0 * aw);
      g[1] = ((y0 + y1) * 0.5f - acy) / (VAR0 * ah);
      g[2] = logf((x1 - x0) / aw) / VAR1;
      g[3] = logf((y1 - y0) / ah) / VAR1;
      const float* pl = pred_loc + idx * 4;
      for (int i = 0; i < 4; ++i) {
        float d = fabsf(pl[i] - g[i]);
        lsum += (d < 1.f) ? 0.5f * d * d : d - 0.5f;
      }
    }
  }
  fred[tid] = lsum;
  ired[tid] = lpos;
  __syncthreads();
  for (int s = 128; s > 0; s >>= 1) {
    if (tid < s) { fred[tid] += fred[tid + s]; ired[tid] += ired[tid + s]; }
    __syncthreads();
  }
  if (tid == 0) {
    loc_part[blockIdx.y * gridDim.x + blockIdx.x] = fred[0];
    atomicAdd(&pos_num[b], ired[0]);  // integer atomic: exact & order-independent
  }
}

// ---------------- K4: CE with WMMA row-sum (16 rows per wave) ----------------
// A-matrix 16x4 f32 layout: lane i (i<16) holds A[i][0..1], lane i+16 holds A[i][2..3].
// B = all-ones 4x16  =>  D[i][*] = sum_k A[i][k]  (exact f32 FMA on the matrix pipe).
// Fast __expf/__logf keep the VALU cost ~2 ops/element so the kernel stays at the
// 255MB/23.3TB/s ~= 11us HBM roofline instead of going VALU-bound on precise expf.
__global__ void k_ce(const float* __restrict__ pred_conf, const int* __restrict__ conf_t,
                     float* __restrict__ ce, float* __restrict__ ce_part) {
  int tid = threadIdx.x;
  int lane = tid & 31;
  int wave = blockIdx.x * (blockDim.x >> 5) + (tid >> 5);
  int base = wave * 16;               // 16 consecutive (b,a) rows per wave
  int rlane = lane & 15;              // row this lane feeds
  int row = base + rlane;
  int koff = (lane & 16) ? 2 : 0;     // class sub-offset within each chunk of 4
  const float* rp = pred_conf + (size_t)row * NC + koff;

  __builtin_prefetch(pred_conf + (size_t)(base + 16) * NC, 0, 3);

  float va[20], vb[20], v80;
  float mx = -3.4e38f;
#pragma unroll
  for (int c = 0; c < 20; ++c) {       // classes 4c+koff, 4c+koff+1  (covers 0..79)
    va[c] = rp[c * 4 + 0];
    vb[c] = rp[c * 4 + 1];
    mx = fmaxf(mx, fmaxf(va[c], vb[c]));
  }
  if (!(lane & 16)) { v80 = rp[80]; mx = fmaxf(mx, v80); } else v80 = 0.f;
  mx = fmaxf(mx, __shfl_xor(mx, 16, 32));   // full row max in lanes i and i+16

  v8f acc = {0.f, 0.f, 0.f, 0.f, 0.f, 0.f, 0.f, 0.f};
  v2f ones; ones.x = 1.f; ones.y = 1.f;
#pragma unroll
  for (int c = 0; c < 20; ++c) {
    v2f aa;
    aa.x = __expf(va[c] - mx);          // v_mul + v_exp_f32 (TRANS pipe)
    aa.y = __expf(vb[c] - mx);
    acc = __builtin_amdgcn_wmma_f32_16x16x4_f32(false, aa, false, ones, (short)0, acc,
                                                false, false);
  }
  {  // chunk 20: only class 80 (lane<16, k=0) is valid
    v2f aa;
    aa.x = (lane & 16) ? 0.f : __expf(v80 - mx);
    aa.y = 0.f;
    acc = __builtin_amdgcn_wmma_f32_16x16x4_f32(false, aa, false, ones, (short)0, acc,
                                                false, false);
  }

  // D layout: VGPR r, lanes 0-15 -> M=r ; lanes 16-31 -> M=8+r
  int r7 = lane & 7;
  float s = (r7 & 4) ? ((r7 & 2) ? ((r7 & 1) ? acc[7] : acc[6]) : ((r7 & 1) ? acc[5] : acc[4]))
                     : ((r7 & 2) ? ((r7 & 1) ? acc[3] : acc[2]) : ((r7 & 1) ? acc[1] : acc[0]));
  float mx_o = __shfl_xor(mx, 24, 32);          // lanes 16..23 fetch rows 8..15 max
  float mxr = (lane & 16) ? mx_o : mx;
  bool writer = ((lane & 8) == 0);              // lanes 0-7 -> rows 0-7; 16-23 -> rows 8-15
  int row_local = r7 + ((lane & 16) ? 8 : 0);

  float val = 0.f;
  if (writer) {
    int r = base + row_local;
    int cls = conf_t[r];
    float x = pred_conf[(size_t)r * NC + cls];
    float cev = -(x - mxr - __logf(s));
    ce[r] = cev;
    if (cls > 0) val = cev;                     // positive CE contribution
  }
  for (int off = 16; off > 0; off >>= 1) val += __shfl_xor(val, off, 32);
  if (lane == 0) ce_part[wave] = val;
}

// ---------------- K5: hard-negative mining (exact top-k sum via LDS radix search) ---
__global__ void k_negmine(const float* __restrict__ ce, const int* __restrict__ conf_t,
                          const int* __restrict__ pos_num, float* __restrict__ neg_sum) {
  __shared__ unsigned skey[24576];  // 96 KB of the 320 KB WGP LDS
  __shared__ unsigned ured[256];
  __shared__ float fred[256];
  int b = blockIdx.x, tid = threadIdx.x;
  const float* cb = ce + (size_t)b * NA;
  const int* tb = conf_t + (size_t)b * NA;
  for (int i = tid; i < NA; i += 256) {
    skey[i] = (tb[i] == 0) ? f2k(cb[i]) : 0u;   // 0 = excluded (valid keys >= 0x80000000)
  }
  __syncthreads();
  int p = pos_num[b];
  int k = max(10, min(3 * p, NA - p));
  unsigned T = 0;
  for (int bit = 31; bit >= 0; --bit) {   // largest T with count(key>=T) >= k
    unsigned cand = T | (1u << bit);
    unsigned cnt = 0;
    for (int i = tid; i < NA; i += 256) {
      unsigned kk = skey[i];
      cnt += (kk != 0u && kk >= cand) ? 1u : 0u;
    }
    ured[tid] = cnt;
    __syncthreads();
    for (int s = 128; s > 0; s >>= 1) {
      if (tid < s) ured[tid] += ured[tid + s];
      __syncthreads();
    }
    unsigned total = ured[0];
    __syncthreads();
    if (total >= (unsigned)k) T = cand;
  }
  unsigned cg = 0, ceq = 0;
  float sg = 0.f;
  for (int i = tid; i < NA; i += 256) {
    unsigned kk = skey[i];
    if (kk == 0u) continue;
    if (kk > T) { cg++; sg += k2f(kk); }
    else if (kk == T) ceq++;
  }
  ured[tid] = cg; fred[tid] = sg;
  __syncthreads();
  for (int s = 128; s > 0; s >>= 1) {
    if (tid < s) { ured[tid] += ured[tid + s]; fred[tid] += fred[tid + s]; }
    __syncthreads();
  }
  unsigned cg_t = ured[0];
  float sg_t = fred[0];
  __syncthreads();
  ured[tid] = ceq;
  __syncthreads();
  for (int s = 128; s > 0; s >>= 1) {
    if (tid < s) ured[tid] += ured[tid + s];
    __syncthreads();
  }
  unsigned ceq_t = ured[0];
  if (tid == 0) {
    float r = sg_t;
    int extra = k - (int)cg_t;
    if (extra > 0) {
      int take = min(extra, (int)ceq_t);
      if (take > 0) r += (float)take * k2f(T);
    }
    neg_sum[b] = r;
  }
}

// ---------------- K6: deterministic final reduction ----------------
__global__ void k_final(const float* __restrict__ loc_part, int nloc,
                        const float* __restrict__ ce_part, int nce,
                        const int* __restrict__ pos_num, const float* __restrict__ neg_sum,
                        float* __restrict__ out) {
  __shared__ float fred[256];
  __shared__ int ired[256];
  int tid = threadIdx.x;
  float ls = 0.f; for (int i = tid; i < nloc; i += 256) ls += loc_part[i];
  float cs = 0.f; for (int i = tid; i < nce; i += 256) cs += ce_part[i];
  int pn = 0;     for (int i = tid; i < NB; i += 256) pn += pos_num[i];
  float ns = 0.f; for (int i = tid; i < NB; i += 256) ns += neg_sum[i];
  fred[tid] = ls; ired[tid] = pn;
  __syncthreads();
  for (int s = 128; s > 0; s >>= 1) {
    if (tid < s) { fred[tid] += fred[tid + s]; ired[tid] += ired[tid + s]; }
    __syncthreads();
  }
  float loc_t = fred[0]; int N = ired[0];
  __syncthreads();
  fred[tid] = cs; __syncthreads();
  for (int s = 128; s > 0; s >>= 1) { if (tid < s) fred[tid] += fred[tid + s]; __syncthreads(); }
  float posce = fred[0];
  __syncthreads();
  fred[tid] = ns; __syncthreads();
  for (int s = 128; s > 0; s >>= 1) { if (tid < s) fred[tid] += fred[tid + s]; __syncthreads(); }
  float negt = fred[0];
  if (tid == 0) {
    float Nf = (float)N;
    out[0] = (posce + negt) / Nf;
    out[1] = loc_t / Nf;
    out[2] = Nf;
  }
}

// ---------------- host glue ----------------
extern "C" void kernel_launch(void* const* d_in, const int* in_sizes, int n_in,
                              void* d_out, int out_size, void* d_ws, size_t ws_size,
                              hipStream_t stream) {
  (void)in_sizes; (void)n_in; (void)out_size; (void)ws_size;
  const float* pred_conf = (const float*)d_in[0];
  const float* pred_loc  = (const float*)d_in[1];
  const float* anchors   = (const float*)d_in[2];
  const float* targets   = (const float*)d_in[3];
  float* out = (float*)d_out;

  char* ws = (char*)d_ws;
  size_t o = 0;
  auto take = [&](size_t bytes) { size_t r = o; o = (o + bytes + 255) & ~(size_t)255; return r; };
  const size_t nBA = (size_t)NB * NA;
  int*                conf_t   = (int*)(ws + take(nBA * 4));
  float*              ce_arr   = (float*)(ws + take(nBA * 4));
  float*              best_iou = (float*)(ws + take(nBA * 4));
  int*                best_idx = (int*)(ws + take(nBA * 4));
  unsigned long long* packed   = (unsigned long long*)(ws + take((size_t)NB * NM * 8));
  int*                pos_num  = (int*)(ws + take(NB * 4));
  const int ablocks = (NA + 255) / 256;               // 96
  const int nloc = ablocks * NB;                      // 3072
  float*              loc_part = (float*)(ws + take((size_t)nloc * 4));
  const int nwaves = (int)(nBA / 16);                 // 49128
  float*              ce_part  = (float*)(ws + take((size_t)nwaves * 4));
  float*              neg_sum  = (float*)(ws + take(NB * 4));

  k_init<<<3, 256, 0, stream>>>(packed, pos_num);
  {
    dim3 g(ablocks, NB);
    k_match<<<g, 256, 0, stream>>>(anchors, targets, best_iou, best_idx, packed);
  }
  k_guarantee<<<1, 32, 0, stream>>>(packed, best_iou, best_idx);
  {
    dim3 g(ablocks, NB);
    k_encode<<<g, 256, 0, stream>>>(targets, anchors, pred_loc, best_iou, best_idx,
                                    conf_t, loc_part, pos_num);
  }
  {
    int blocks = nwaves / 8;  // 8 waves (256 threads) per block
    k_ce<<<blocks, 256, 0, stream>>>(pred_conf, conf_t, ce_arr, ce_part);
  }
  k_negmine<<<NB, 256, 0, stream>>>(ce_arr, conf_t, pos_num, neg_sum);
  k_final<<<1, 256, 0, stream>>>(loc_part, nloc, ce_part, nwaves, pos_num, neg_sum, out);
}